// DenoiseNet_6502580486604
// MI455X (gfx1250) — compile-verified
//
#include <hip/hip_runtime.h>
#include <hip/hip_bf16.h>

typedef __attribute__((ext_vector_type(2))) float v2f;
typedef __attribute__((ext_vector_type(8))) float v8f;

#define B_   8
#define N_   8192
#define M_   8192
#define P_   128
#define K1_  32
#define K2_  4
#define FD_  128
#define H_   128
#define BIGF 3.0e38f

// ---------------------------------------------------------------- zero output
__global__ void k_zero(float* out) {
    if (threadIdx.x == 0 && blockIdx.x == 0) out[0] = 0.0f;
}

// ------------------------------------------------- feat[b,p,:] = pt @ W_feat
__global__ void __launch_bounds__(FD_) k_feat(const float* __restrict__ noisy,
                                              const int*   __restrict__ sidx,
                                              const float* __restrict__ Wf,
                                              float*       __restrict__ feat) {
    int bp = blockIdx.x;
    int b = bp / P_, p = bp % P_;
    int si = sidx[p];
    const float* pt = noisy + ((size_t)b * N_ + si) * 3;
    float x = pt[0], y = pt[1], z = pt[2];
    int h = threadIdx.x;
    feat[(size_t)bp * FD_ + h] = x * Wf[0 * FD_ + h] + y * Wf[1 * FD_ + h] + z * Wf[2 * FD_ + h];
}

// ----------------------------- KNN1: top-32 neighbors of q in noisy cloud.
__global__ void __launch_bounds__(256) k_knn1(const float* __restrict__ noisy,
                                              const int*   __restrict__ sidx,
                                              float*       __restrict__ fbuf) {
    __shared__ float d2[N_];
    __shared__ float rd[256];
    __shared__ int   ri[256];
    int bp = blockIdx.x;
    int b = bp / P_, p = bp % P_;
    int t = threadIdx.x;
    int si = sidx[p];
    const float* base = noisy + (size_t)b * N_ * 3;
    float qx = base[si * 3 + 0], qy = base[si * 3 + 1], qz = base[si * 3 + 2];
    for (int r = t; r < N_; r += 256) {
        float dx = base[r * 3 + 0] - qx;
        float dy = base[r * 3 + 1] - qy;
        float dz = base[r * 3 + 2] - qz;
        d2[r] = dx * dx + dy * dy + dz * dz;
    }
    __syncthreads();
    for (int k = 0; k < K1_; ++k) {
        float bd = BIGF; int bi = 0;
        for (int r = t; r < N_; r += 256) {
            float v = d2[r];
            if (v < bd) { bd = v; bi = r; }
        }
        rd[t] = bd; ri[t] = bi;
        __syncthreads();
        for (int s = 128; s > 0; s >>= 1) {
            if (t < s) {
                float a = rd[t], bb = rd[t + s];
                int ia = ri[t], ib = ri[t + s];
                if (bb < a || (bb == a && ib < ia)) { rd[t] = bb; ri[t] = ib; }
            }
            __syncthreads();
        }
        int wi = ri[0];
        if (t == 0) {
            float* o = fbuf + ((size_t)bp * K1_ + k) * 3;
            o[0] = base[wi * 3 + 0];
            o[1] = base[wi * 3 + 1];
            o[2] = base[wi * 3 + 2];
            d2[wi] = BIGF;
        }
        __syncthreads();
    }
}

// ----------------------------- KNN2 + ground score.
// WMMA f32 16x16x4 distance tiles; clean cloud staged through LDS with
// double-buffered global_load_async_to_lds_b128 (ASYNCcnt); branchless top-4.
#define CHUNK_PTS 1024
#define CHUNK_FLT (CHUNK_PTS * 3)        // 3072 floats = 12 KB
#define NCHUNK    (M_ / CHUNK_PTS)       // 8

__global__ void __launch_bounds__(256) k_knn2(const float* __restrict__ clean,
                                              const float* __restrict__ fbuf,
                                              float*       __restrict__ ground) {
    __shared__ __align__(16) float spts[2][CHUNK_FLT];
    int t    = threadIdx.x;
    int lane = t & 31;
    int wave = t >> 5;
    int qbase = (blockIdx.x * 8 + wave) * 16;
    int n  = lane & 15;
    int gq = qbase + n;                 // global query-point id in [0, B*P*K1)
    int b  = (blockIdx.x * 128) >> 12;  // uniform per block (128 | 4096)
    const float* cb = clean + (size_t)b * M_ * 3;
    const char*  gsrc = (const char*)cb;
    const float* fp = fbuf + (size_t)gq * 3;
    float fx = fp[0], fy = fp[1], fz = fp[2];

    // B operand (4 x 16 queries): lanes 0-15 supply K=0,1; lanes 16-31 K=2,3
    v2f bm;
    bm.x = (lane < 16) ? (-2.0f * fx) : (-2.0f * fz);
    bm.y = (lane < 16) ? (-2.0f * fy) : 1.0f;

    float bd0 = BIGF, bd1 = BIGF, bd2 = BIGF, bd3 = BIGF;
    int   bi0 = 0, bi1 = 0, bi2 = 0, bi3 = 0;
    int mofs = (lane >= 16) ? 8 : 0;

    // async-copy one 12KB chunk (3 x b128 per thread, fully coalesced)
    auto issue_chunk = [&](int c, int buf) {
        unsigned lbase = (unsigned)(uintptr_t)&spts[buf][0];
        const char* gbase = gsrc + (size_t)c * (CHUNK_FLT * 4);
#pragma unroll
        for (int j = 0; j < 3; ++j) {
            unsigned    loff = lbase + (unsigned)(j * 4096 + t * 16);
            const char* ga   = gbase + (size_t)(j * 4096 + t * 16);
            asm volatile("global_load_async_to_lds_b128 %0, %1, off"
                         :: "v"(loff), "v"(ga) : "memory");
        }
    };

    issue_chunk(0, 0);
    asm volatile("s_wait_asynccnt 0" ::: "memory");
    __syncthreads();

    for (int ch = 0; ch < NCHUNK; ++ch) {
        int buf = ch & 1;
        if (ch + 1 < NCHUNK) issue_chunk(ch + 1, buf ^ 1);  // overlap DMA w/ compute
        const float* sp = &spts[buf][0];
#pragma unroll 4
        for (int tl = 0; tl < CHUNK_PTS / 16; ++tl) {
            int rb = ch * CHUNK_PTS + tl * 16;
            const float* rp = sp + (tl * 16 + n) * 3;
            float rx = rp[0], ry = rp[1], rz = rp[2];
            float rr = rx * rx + ry * ry + rz * rz;
            v2f a;
            a.x = (lane < 16) ? rx : rz;
            a.y = (lane < 16) ? ry : rr;
            v8f c = {0.f, 0.f, 0.f, 0.f, 0.f, 0.f, 0.f, 0.f};
            c = __builtin_amdgcn_wmma_f32_16x16x4_f32(
                    false, a, false, bm, (short)0, c, false, false);
#pragma unroll
            for (int v = 0; v < 8; ++v) {
                float d = c[v];
                int ridx = rb + v + mofs;
                // branchless sorted insert into ascending top-4
                bool l0 = d < bd0, l1 = d < bd1, l2 = d < bd2, l3 = d < bd3;
                float t0 = bd0, t1 = bd1, t2 = bd2;
                int   u0 = bi0, u1 = bi1, u2 = bi2;
                bd0 = l0 ? d : bd0;              bi0 = l0 ? ridx : bi0;
                bd1 = l0 ? t0 : (l1 ? d : bd1);  bi1 = l0 ? u0 : (l1 ? ridx : bi1);
                bd2 = l1 ? t1 : (l2 ? d : bd2);  bi2 = l1 ? u1 : (l2 ? ridx : bi2);
                bd3 = l2 ? t2 : (l3 ? d : bd3);  bi3 = l2 ? u2 : (l3 ? ridx : bi3);
            }
        }
        asm volatile("s_wait_asynccnt 0" ::: "memory");
        __syncthreads();
    }

    // merge sorted top-4 lists of lane L and lane L^16
    float ad[4] = {bd0, bd1, bd2, bd3};
    int   ai[4] = {bi0, bi1, bi2, bi3};
    float od[4]; int oi[4];
#pragma unroll
    for (int j = 0; j < 4; ++j) {
        od[j] = __shfl(ad[j], lane ^ 16, 32);
        oi[j] = __shfl(ai[j], lane ^ 16, 32);
    }
    int mi[4];
    int i0 = 0, j0 = 0;
#pragma unroll
    for (int tsel = 0; tsel < 4; ++tsel) {
        bool takeA = (ad[i0] < od[j0]) || (ad[i0] == od[j0] && ai[i0] < oi[j0]);
        mi[tsel] = takeA ? ai[i0] : oi[j0];
        if (takeA) ++i0; else ++j0;
    }

    if (lane < 16) {
        float sx = 0.f, sy = 0.f, sz = 0.f;
#pragma unroll
        for (int j = 0; j < 4; ++j) {
            const float* np = cb + (size_t)mi[j] * 3;
            sx += np[0]; sy += np[1]; sz += np[2];
        }
        float* g = ground + (size_t)gq * 3;
        g[0] = sx * 0.25f - fx;
        g[1] = sy * 0.25f - fy;
        g[2] = sz * 0.25f - fz;
    }
}

// ----------------------------- bias = feat @ W2  (1024x128 @ 128x128), WMMA.
// One wave per 16x16 output tile, K=128 as 32 chained 16x16x4 WMMAs.
__global__ void __launch_bounds__(256) k_bias(const float* __restrict__ feat,
                                              const float* __restrict__ W2,
                                              float*       __restrict__ bias) {
    int t = threadIdx.x, lane = t & 31, wave = t >> 5;
    int gwave = blockIdx.x * 8 + wave;    // 0..511
    int mt = gwave >> 3, nt = gwave & 7;  // 64 row tiles x 8 col tiles
    int n = lane & 15;
    int koff = (lane < 16) ? 0 : 2;
    const float* arow = feat + (size_t)(mt * 16 + n) * FD_;
    const float* bcol = W2 + nt * 16 + n;
    v8f c = {0.f, 0.f, 0.f, 0.f, 0.f, 0.f, 0.f, 0.f};
    for (int k0 = 0; k0 < FD_; k0 += 4) {
        v2f a, bmat;
        a.x    = arow[k0 + koff];
        a.y    = arow[k0 + koff + 1];
        bmat.x = bcol[(size_t)(k0 + koff) * H_];
        bmat.y = bcol[(size_t)(k0 + koff + 1) * H_];
        c = __builtin_amdgcn_wmma_f32_16x16x4_f32(
                false, a, false, bmat, (short)0, c, false, false);
    }
    int mofs = (lane >= 16) ? 8 : 0;
#pragma unroll
    for (int v = 0; v < 8; ++v)
        bias[(size_t)(mt * 16 + v + mofs) * H_ + nt * 16 + n] = c[v];
}

// ----------------------------- MLP + loss. One workgroup per (b,p).
__global__ void __launch_bounds__(H_) k_mlp(const float* __restrict__ noisy,
                                            const int*   __restrict__ sidx,
                                            const float* __restrict__ W1,
                                            const float* __restrict__ W3,
                                            const float* __restrict__ biasbuf,
                                            const float* __restrict__ fbuf,
                                            const float* __restrict__ ground,
                                            float*       __restrict__ out) {
    __shared__ float red[3][H_];
    int bp = blockIdx.x;
    int b = bp / P_, p = bp % P_;
    int h = threadIdx.x;
    int si = sidx[p];
    const float* qp = noisy + ((size_t)b * N_ + si) * 3;
    float qx = qp[0], qy = qp[1], qz = qp[2];
    float bias = biasbuf[(size_t)bp * H_ + h];
    float w10 = W1[0 * H_ + h], w11 = W1[1 * H_ + h], w12 = W1[2 * H_ + h];
    float w30 = W3[h * 3 + 0], w31 = W3[h * 3 + 1], w32 = W3[h * 3 + 2];
    float acc = 0.f;
    for (int k = 0; k < K1_; ++k) {
        const float* fk = fbuf + ((size_t)bp * K1_ + k) * 3;
        float x0 = fk[0] - qx, x1 = fk[1] - qy, x2 = fk[2] - qz;
        float ht = x0 * w10 + x1 * w11 + x2 * w12 + bias;
        ht = ht > 0.f ? ht : 0.f;
        red[0][h] = ht * w30;
        red[1][h] = ht * w31;
        red[2][h] = ht * w32;
        __syncthreads();
        for (int s = 64; s > 0; s >>= 1) {
            if (h < s) {
                red[0][h] += red[0][h + s];
                red[1][h] += red[1][h + s];
                red[2][h] += red[2][h + s];
            }
            __syncthreads();
        }
        if (h == 0) {
            const float* g = ground + ((size_t)bp * K1_ + k) * 3;
            float d0 = red[0][0] - g[0];
            float d1 = red[1][0] - g[1];
            float d2v = red[2][0] - g[2];
            acc += d0 * d0 + d1 * d1 + d2v * d2v;
        }
        __syncthreads();
    }
    if (h == 0) atomicAdd(out, acc * (50.0f / 32768.0f)); // mean over B*P*K1
}

extern "C" void kernel_launch(void* const* d_in, const int* in_sizes, int n_in,
                              void* d_out, int out_size, void* d_ws, size_t ws_size,
                              hipStream_t stream) {
    const float* noisy = (const float*)d_in[0];
    const float* clean = (const float*)d_in[1];
    const int*   sidx  = (const int*)d_in[2];
    const float* Wf    = (const float*)d_in[3];
    const float* W1    = (const float*)d_in[4];
    const float* W2    = (const float*)d_in[5];
    const float* W3    = (const float*)d_in[6];
    float* out = (float*)d_out;

    float* ws     = (float*)d_ws;
    float* feat   = ws;                                     // B*P*FD
    float* fbuf   = feat + (size_t)B_ * P_ * FD_;           // B*P*K1*3
    float* ground = fbuf + (size_t)B_ * P_ * K1_ * 3;       // B*P*K1*3
    float* bias   = ground + (size_t)B_ * P_ * K1_ * 3;     // B*P*H

    hipLaunchKernelGGL(k_zero, dim3(1), dim3(1), 0, stream, out);
    hipLaunchKernelGGL(k_feat, dim3(B_ * P_), dim3(FD_), 0, stream, noisy, sidx, Wf, feat);
    hipLaunchKernelGGL(k_knn1, dim3(B_ * P_), dim3(256), 0, stream, noisy, sidx, fbuf);
    hipLaunchKernelGGL(k_knn2, dim3((B_ * P_ * K1_) / (16 * 8)), dim3(256), 0, stream,
                       clean, fbuf, ground);
    hipLaunchKernelGGL(k_bias, dim3((B_ * P_ / 16) * (H_ / 16) / 8), dim3(256), 0, stream,
                       feat, W2, bias);
    hipLaunchKernelGGL(k_mlp, dim3(B_ * P_), dim3(H_), 0, stream,
                       noisy, sidx, W1, W3, bias, fbuf, ground, out);
}